// DifferentiableFloorPlan_39608188403885
// MI455X (gfx1250) — compile-verified
//
#include <hip/hip_runtime.h>
#include <hip/hip_bf16.h>

typedef __attribute__((ext_vector_type(2))) float v2f;
typedef __attribute__((ext_vector_type(4))) float v4f;
typedef __attribute__((ext_vector_type(8))) float v8f;

#define RES 1024
#define NUM_ROOMS 8
#define LAYOUT_ELEMS (NUM_ROOMS * RES * RES)   // 8388608
#define FLOW_ELEMS   (RES * RES)               // 1048576

// ---------------------------------------------------------------------------
// Kernel 1: zero the flow-field accumulation region + the max scratch slot.
// ---------------------------------------------------------------------------
__global__ void fp_zero_kernel(float* __restrict__ flow, unsigned* __restrict__ wsmax) {
    int i = blockIdx.x * blockDim.x + threadIdx.x;
    if (i < FLOW_ELEMS) flow[i] = 0.0f;
    if (i == 0) *wsmax = 0u;
}

// ---------------------------------------------------------------------------
// Kernel 2: dynamic_layout via rank-1 WMMA.
// One wave computes one 16x16 tile of Ex (.) Ey with V_WMMA_F32_16X16X4_F32:
//   A (16x4 f32): K=0 column = Ex tile (VGPR0 lanes 0-15), rest zero.
//   B (4x16 f32): K=0 row    = Ey tile (VGPR0 lanes 0-15), rest zero.
//   D = A*B = Ex (.) Ey, then multiply by (1 - wall) and store.
// Wall (4 MB) is re-read 8x -> regular-temporal loads (L2 resident).
// Output (33.5 MB) is write-once -> non-temporal stores.
// 4096 tiles/room * 8 rooms = 32768 waves = 4096 blocks of 256 threads.
// ---------------------------------------------------------------------------
__global__ void fp_layout_kernel(const float* __restrict__ rp,
                                 const float* __restrict__ wall,
                                 float* __restrict__ out) {
    const int lane = threadIdx.x & 31;
    const int wave = threadIdx.x >> 5;
    const int tileLinear = blockIdx.x * 8 + wave;       // 0 .. 32767
    const int room = tileLinear >> 12;                  // 4096 tiles per room
    const int t    = tileLinear & 4095;
    const int ti   = t >> 6;                            // tile row (0..63)
    const int tj   = t & 63;                            // tile col (0..63)

    const float cx = rp[room * 4 + 0];
    const float cy = rp[room * 4 + 1];
    const float sx = rp[room * 4 + 2];
    const float sy = rp[room * 4 + 3];
    const float inv2sx2 = 0.5f / (sx * sx);
    const float inv2sy2 = 0.5f / (sy * sy);

    const int l15 = lane & 15;
    // linspace(0,1,1024): lin[i] = i / 1023
    const float xi = (float)(ti * 16 + l15) * (1.0f / 1023.0f);
    const float yj = (float)(tj * 16 + l15) * (1.0f / 1023.0f);
    const float dx = xi - cx;
    const float dy = yj - cy;
    const float exv = (lane < 16) ? __expf(-dx * dx * inv2sx2) : 0.0f;
    const float eyv = (lane < 16) ? __expf(-dy * dy * inv2sy2) : 0.0f;

    v2f a; a.x = exv; a.y = 0.0f;   // K=0 col = Ex; K=1..3 zero
    v2f b; b.x = eyv; b.y = 0.0f;   // K=0 row = Ey; K=1..3 zero
    v8f c = {};
    // D = A x B + 0  -> 16x16 outer product tile
    v8f d = __builtin_amdgcn_wmma_f32_16x16x4_f32(
        /*neg_a=*/false, a, /*neg_b=*/false, b,
        /*c_mod=*/(short)0, c, /*reuse_a=*/false, /*reuse_b=*/false);

    // D layout: VGPR v, lanes 0-15: M=v,  N=lane; lanes 16-31: M=v+8, N=lane-16
    const int n     = tj * 16 + l15;
    const int mbase = ti * 16 + ((lane >= 16) ? 8 : 0);
    float* outp = out + (size_t)room * (RES * RES);

    __builtin_prefetch(&wall[(size_t)mbase * RES + n], 0, 1);

#pragma unroll
    for (int v = 0; v < 8; ++v) {
        const int i = mbase + v;
        const float w = wall[(size_t)i * RES + n];          // temporal: keep in L2
        __builtin_nontemporal_store(d[v] * (1.0f - w),
                                    &outp[(size_t)i * RES + n]);  // streamed out
    }
}

// ---------------------------------------------------------------------------
// Kernel 3: agent histogram. Two agents per thread via one b128 load
// (non-temporal: 40 MB read-once stream), float atomics into flow region.
// ---------------------------------------------------------------------------
__global__ void fp_hist_kernel(const v4f* __restrict__ pos2,
                               float* __restrict__ flow, int n_agents) {
    const int idx = blockIdx.x * blockDim.x + threadIdx.x;   // pair index
    const int base = idx * 2;
    if (base >= n_agents) return;
    const v4f p = __builtin_nontemporal_load(&pos2[idx]);    // agents 2i, 2i+1

    int ix0 = (int)(p.x * (float)RES);  // trunc toward zero == astype(int32)
    int iy0 = (int)(p.y * (float)RES);
    ix0 = min(max(ix0, 0), RES - 1);
    iy0 = min(max(iy0, 0), RES - 1);
    atomicAdd(&flow[ix0 * RES + iy0], 1.0f);

    if (base + 1 < n_agents) {
        int ix1 = (int)(p.z * (float)RES);
        int iy1 = (int)(p.w * (float)RES);
        ix1 = min(max(ix1, 0), RES - 1);
        iy1 = min(max(iy1, 0), RES - 1);
        atomicAdd(&flow[ix1 * RES + iy1], 1.0f);
    }
}

// ---------------------------------------------------------------------------
// Kernel 4: global max of flow counts (non-negative floats -> uint bit compare).
// ---------------------------------------------------------------------------
__global__ void fp_max_kernel(const float* __restrict__ flow,
                              unsigned* __restrict__ wsmax) {
    float m = 0.0f;
    for (int i = blockIdx.x * blockDim.x + threadIdx.x; i < FLOW_ELEMS;
         i += gridDim.x * blockDim.x) {
        m = fmaxf(m, flow[i]);
    }
    // wave32 reduction
    for (int off = 16; off > 0; off >>= 1)
        m = fmaxf(m, __shfl_down(m, off, 32));
    __shared__ float smax[8];
    if ((threadIdx.x & 31) == 0) smax[threadIdx.x >> 5] = m;
    __syncthreads();
    if (threadIdx.x == 0) {
        float bm = smax[0];
#pragma unroll
        for (int w = 1; w < 8; ++w) bm = fmaxf(bm, smax[w]);
        atomicMax(wsmax, __float_as_uint(bm));
    }
}

// ---------------------------------------------------------------------------
// Kernel 5: normalize flow in place: flow /= (max + 1e-6).
// ---------------------------------------------------------------------------
__global__ void fp_norm_kernel(float* __restrict__ flow,
                               const unsigned* __restrict__ wsmax) {
    int i = blockIdx.x * blockDim.x + threadIdx.x;
    if (i >= FLOW_ELEMS) return;
    const float mx = __uint_as_float(*wsmax);
    __builtin_nontemporal_store(flow[i] / (mx + 1e-6f), &flow[i]);
}

extern "C" void kernel_launch(void* const* d_in, const int* in_sizes, int n_in,
                              void* d_out, int out_size, void* d_ws, size_t ws_size,
                              hipStream_t stream) {
    const float*  agent_pos = (const float*)d_in[0];   // (N_AGENTS, 2)
    const float*  room_par  = (const float*)d_in[1];   // (8, 4)
    const float*  wall      = (const float*)d_in[2];   // (1024, 1024)
    const int n_agents = in_sizes[0] / 2;

    float* layout_out = (float*)d_out;                       // [0, 8*1024*1024)
    float* flow_out   = (float*)d_out + LAYOUT_ELEMS;        // [.., +1024*1024)
    unsigned* wsmax   = (unsigned*)d_ws;                     // 4 bytes scratch

    // 1. zero flow accumulator + max slot
    fp_zero_kernel<<<(FLOW_ELEMS + 255) / 256, 256, 0, stream>>>(flow_out, wsmax);

    // 2. dynamic_layout via WMMA rank-1 outer product (32768 waves)
    fp_layout_kernel<<<4096, 256, 0, stream>>>(room_par, wall, layout_out);

    // 3. histogram of agent positions (2 agents/thread, b128 NT loads)
    const int n_pairs = (n_agents + 1) / 2;
    fp_hist_kernel<<<(n_pairs + 255) / 256, 256, 0, stream>>>(
        (const v4f*)agent_pos, flow_out, n_agents);

    // 4. global max reduction
    fp_max_kernel<<<1024, 256, 0, stream>>>(flow_out, wsmax);

    // 5. normalize in place
    fp_norm_kernel<<<(FLOW_ELEMS + 255) / 256, 256, 0, stream>>>(flow_out, wsmax);
}